// Analyzer2_55241869361649
// MI455X (gfx1250) — compile-verified
//
#include <hip/hip_runtime.h>
#include <stdint.h>

typedef float v2f __attribute__((ext_vector_type(2)));
typedef float v8f __attribute__((ext_vector_type(8)));

#define NPTS 4096
#define NW   128           // bitmask words per row (4096/32)
#define THRESH 0.5f

// d_out layout (floats, concatenated in reference return order)
#define PC_OFF    0
#define TC_OFF    98304
#define KEEPP_OFF 196608
#define KEEPT_OFF 229376
#define TP_OFF    262144
#define FP_OFF    294912
#define FN_OFF    327680

// d_ws layout (bytes)
#define WS_CONFP    0
#define WS_CONFT    (NPTS*8*4)                 // 131072
#define WS_KEEPBITS (2*NPTS*8*4)               // 262144
#define WS_MAT      (WS_KEEPBITS + 16*NW*4)    // 270336
#define MAT_WORDS   ((size_t)NPTS*NW)          // words per problem matrix

// ---------------- Kernel B: transform + bitonic argsort (desc, stable) -----
__global__ __launch_bounds__(256) void k_transform_sort(
    const float* __restrict__ pred, const float* __restrict__ targ,
    float* __restrict__ out, float* __restrict__ confP, float* __restrict__ confT)
{
  __shared__ unsigned long long keys[NPTS];     // 32 KB LDS
  const int blk = blockIdx.x;                   // 0..15
  const int src = blk >> 3;                     // 0=P, 1=T
  const int p   = blk & 7;                      // b*2+e
  const int bb  = p >> 1, e = p & 1;
  const float* in = (src == 0) ? pred : targ;
  const int tid = threadIdx.x;

  for (int n = tid; n < NPTS; n += 256) {
    int z = n >> 10, xg = (n >> 5) & 31, yg = n & 31;
    int base = (((bb*32 + xg)*32 + yg)*4 + z)*8 + e*4;
    float conf = in[base + 3];
    unsigned u = __float_as_uint(conf);
    unsigned ord = (u & 0x80000000u) ? ~u : (u | 0x80000000u);  // ascending order key
    keys[n] = ((unsigned long long)(~ord) << 32) | (unsigned)n; // asc sort => desc conf, stable
  }
  __syncthreads();
  for (unsigned k = 2; k <= NPTS; k <<= 1)
    for (unsigned j = k >> 1; j > 0; j >>= 1) {
      for (unsigned i = tid; i < NPTS; i += 256) {
        unsigned ixj = i ^ j;
        if (ixj > i) {
          unsigned long long a = keys[i], b2 = keys[ixj];
          bool up = ((i & k) == 0);
          if ((a > b2) == up) { keys[i] = b2; keys[ixj] = a; }
        }
      }
      __syncthreads();
    }

  float* cOut = out + ((src == 0) ? PC_OFF : TC_OFF) + (size_t)p*NPTS*3;
  float* cw   = ((src == 0) ? confP : confT) + p*NPTS;
  for (int i = tid; i < NPTS; i += 256) {
    unsigned idx = (unsigned)keys[i];
    int z = idx >> 10, xg = (idx >> 5) & 31, yg = idx & 31;
    int base = (((bb*32 + xg)*32 + yg)*4 + z)*8 + e*4;
    float v0 = in[base], v1 = in[base+1], v2 = in[base+2];
    cOut[i*3+0] = (v2 + (float)z ) * 0.75f;     // (v[...,2]+z) * 3/4
    cOut[i*3+1] = (v0 + (float)xg) * 0.78125f;  // (v[...,0]+x) * 25/32
    cOut[i*3+2] = (v1 + (float)yg) * 0.78125f;  // (v[...,1]+y) * 25/32
    cw[i] = in[base + 3];
  }
}

// ------- Kernel C: pairwise-distance bit matrices via V_WMMA_F32_16X16X4 ---
// z = 0..7  : NMS-P  (row=b suppressed, bit a set iff a<b, both valid, d2<thr)
// z = 8..15 : NMS-T  (same over targets)
// z = 16..23: match  (row=P point, bit over T points; valid&valid&d2<thr)
__global__ __launch_bounds__(256) void k_pairbits(
    const float* __restrict__ out, const float* __restrict__ confP,
    const float* __restrict__ confT, unsigned* __restrict__ mats)
{
  const int z    = blockIdx.z;          // 0..23
  const int kind = z >> 3;              // 0 nmsP, 1 nmsT, 2 match
  const int p    = z & 7;
  const int wave = threadIdx.x >> 5;
  const int lane = threadIdx.x & 31;
  const int wordIdx = blockIdx.x * 8 + wave;   // 0..127 -> which 32-col chunk
  const int ia = wordIdx << 5;                 // first column index of the word
  const int jb = blockIdx.y << 4;              // first row index of the tile
  const bool isNms = (kind < 2);
  if (isNms && ia > jb + 15) return;           // entirely above diagonal: never read

  const float* rows = out + ((kind == 1) ? TC_OFF : PC_OFF) + (size_t)p*NPTS*3;
  const float* cols = out + ((kind == 0) ? PC_OFF : TC_OFF) + (size_t)p*NPTS*3;
  const float* rc   = ((kind == 1) ? confT : confP) + p*NPTS;
  const float* cc   = ((kind == 0) ? confP : confT) + p*NPTS;
  const float dd    = (p & 1) ? (0.528f*1.4f) : (0.74f*1.4f);
  const float dthr2 = dd * dd;

  const int mrow = lane & 15;
  float r0 = rows[(jb+mrow)*3+0], r1 = rows[(jb+mrow)*3+1], r2 = rows[(jb+mrow)*3+2];
  float n2r = r0*r0 + r1*r1 + r2*r2;
  int   vR  = (rc[jb+mrow] > THRESH) ? 1 : 0;

  // A (16x4 f32): lanes 0-15 hold K0,K1 of row M=lane; lanes 16-31 hold K2,K3(=0)
  v2f A;
  if (lane < 16) { A.x = r0; A.y = r1; } else { A.x = r2; A.y = 0.f; }

  unsigned myword = 0;
  for (int h = 0; h < 2; ++h) {
    const int ncol = ia + h*16 + mrow;
    float c0 = cols[ncol*3+0], c1 = cols[ncol*3+1], c2 = cols[ncol*3+2];
    float n2c = c0*c0 + c1*c1 + c2*c2;
    int   vC  = (cc[ncol] > THRESH) ? 1 : 0;

    float dotv[8];
#if __has_builtin(__builtin_amdgcn_wmma_f32_16x16x4_f32)
    v2f Bo;
    if (lane < 16) { Bo.x = c0; Bo.y = c1; } else { Bo.x = c2; Bo.y = 0.f; }
    v8f acc = {0.f,0.f,0.f,0.f,0.f,0.f,0.f,0.f};
    v8f D = __builtin_amdgcn_wmma_f32_16x16x4_f32(
        false, A, false, Bo, (short)0, acc, false, false);
#pragma unroll
    for (int r = 0; r < 8; ++r) dotv[r] = D[r];
#else
#pragma unroll
    for (int r = 0; r < 8; ++r) {
      int msrc = r + ((lane & 16) ? 8 : 0);
      dotv[r] = __shfl(r0, msrc)*c0 + __shfl(r1, msrc)*c1 + __shfl(r2, msrc)*c2;
    }
#endif
#pragma unroll
    for (int r = 0; r < 8; ++r) {
      int msrc = r + ((lane & 16) ? 8 : 0);     // local row index for this lane half
      float n2m = __shfl(n2r, msrc);
      int   vM  = __shfl(vR, msrc);
      bool bit = (n2m + n2c - 2.f*dotv[r] < dthr2) && vM && vC;
      if (isNms) bit = bit && (ncol < jb + msrc);
      unsigned ball = (unsigned)__ballot(bit);  // wave32: 32 valid bits
      unsigned lo = ball & 0xFFFFu;             // row r,   tile cols 0..15
      unsigned hi = ball >> 16;                 // row r+8, tile cols 0..15
      if (lane == r)     myword |= (h ? (lo << 16) : lo);
      if (lane == r + 8) myword |= (h ? (hi << 16) : hi);
    }
  }
  if (lane < 16) {
    unsigned* M = mats + (size_t)z * MAT_WORDS;
    M[(size_t)(jb + lane) * NW + wordIdx] = myword;
  }
}

// ------------- Kernel D: serial greedy NMS scan over bit matrix ------------
__global__ __launch_bounds__(128) void k_nms(
    const unsigned* __restrict__ mats, const float* __restrict__ confP,
    const float* __restrict__ confT, float* __restrict__ out,
    unsigned* __restrict__ keepbits)
{
  __shared__ unsigned keepW[NW];
  __shared__ int flagS;
  const int prob = blockIdx.x;          // 0..15  (0..7 P, 8..15 T)
  const int tid = threadIdx.x;
  const unsigned* M = mats + (size_t)prob * MAT_WORDS;
  const float* cf = (prob < 8) ? (confP + prob*NPTS) : (confT + (prob-8)*NPTS);
  float* keepOut = out + ((prob < 8) ? (KEEPP_OFF + prob*NPTS)
                                     : (KEEPT_OFF + (prob-8)*NPTS));
  keepW[tid] = 0;
  if (tid == 0) flagS = 0;
  __syncthreads();
  for (int b = 0; b < NPTS; ++b) {
    const int nw = (b >> 5) + 1;                       // only words with a <= b
    unsigned m = (tid < nw) ? (M[(size_t)b*NW + tid] & keepW[tid]) : 0u;
    if (m) flagS = 1;                                   // benign race (all write 1)
    __syncthreads();
    if (tid == 0) {
      bool sel = (cf[b] > THRESH) && !flagS;
      if (sel) keepW[b >> 5] |= 1u << (b & 31);
      keepOut[b] = sel ? 1.f : 0.f;
      flagS = 0;
    }
    __syncthreads();
  }
  keepbits[prob*NW + tid] = keepW[tid];
}

// ----------- Kernel E: serial greedy bipartite match over bit matrix -------
__global__ __launch_bounds__(128) void k_match(
    const unsigned* __restrict__ mats, const unsigned* __restrict__ keepbits,
    float* __restrict__ out)
{
  __shared__ unsigned selT[NW], keepPW[NW], keepTW[NW];
  __shared__ int jminS;
  const int p = blockIdx.x;            // 0..7
  const int tid = threadIdx.x;
  const unsigned* D = mats + (size_t)(16 + p) * MAT_WORDS;
  selT[tid] = 0;
  keepPW[tid] = keepbits[p*NW + tid];
  keepTW[tid] = keepbits[(8 + p)*NW + tid];
  if (tid == 0) jminS = 0x7FFFFFFF;
  __syncthreads();
  float* tp = out + TP_OFF + p*NPTS;
  float* fp = out + FP_OFF + p*NPTS;
  float* fn = out + FN_OFF + p*NPTS;
  for (int a = 0; a < NPTS; ++a) {
    const bool kp = (keepPW[a >> 5] >> (a & 31)) & 1;
    if (kp) {
      unsigned v = D[(size_t)a*NW + tid] & keepTW[tid] & ~selT[tid];
      if (v) atomicMin(&jminS, tid*32 + __builtin_ctz(v));   // first available target
    }
    __syncthreads();
    if (tid == 0) {
      int j = jminS;
      bool has = kp && (j != 0x7FFFFFFF);
      if (has) selT[j >> 5] |= 1u << (j & 31);
      tp[a] = has ? 1.f : 0.f;
      fp[a] = (kp && !has) ? 1.f : 0.f;
      jminS = 0x7FFFFFFF;
    }
    __syncthreads();
  }
  for (int j2 = tid; j2 < NPTS; j2 += 128) {
    bool kt = (keepTW[j2 >> 5] >> (j2 & 31)) & 1;
    bool st = (selT [j2 >> 5] >> (j2 & 31)) & 1;
    fn[j2] = (kt && !st) ? 1.f : 0.f;
  }
}

// ---------------------------------------------------------------------------
extern "C" void kernel_launch(void* const* d_in, const int* in_sizes, int n_in,
                              void* d_out, int out_size, void* d_ws, size_t ws_size,
                              hipStream_t stream) {
  (void)in_sizes; (void)n_in; (void)out_size; (void)ws_size;
  const float* pred = (const float*)d_in[0];
  const float* targ = (const float*)d_in[1];
  float* out = (float*)d_out;
  char*  ws  = (char*)d_ws;
  float*    confP    = (float*)(ws + WS_CONFP);
  float*    confT    = (float*)(ws + WS_CONFT);
  unsigned* keepbits = (unsigned*)(ws + WS_KEEPBITS);
  unsigned* mats     = (unsigned*)(ws + WS_MAT);   // 24 x 2MB bit matrices (~48MB, L2-resident)

  k_transform_sort<<<dim3(16), dim3(256), 0, stream>>>(pred, targ, out, confP, confT);
  k_pairbits     <<<dim3(16, 256, 24), dim3(256), 0, stream>>>(out, confP, confT, mats);
  k_nms          <<<dim3(16), dim3(128), 0, stream>>>(mats, confP, confT, out, keepbits);
  k_match        <<<dim3(8), dim3(128), 0, stream>>>(mats, keepbits, out);
}